// Encoder_18966575579326
// MI455X (gfx1250) — compile-verified
//
#include <hip/hip_runtime.h>

typedef __attribute__((ext_vector_type(2))) float v2f;
typedef __attribute__((ext_vector_type(8))) float v8f;

#define EMB   32
#define U1    32
#define U2    16
#define SEQ   1024
#define BATCH 512

__device__ __forceinline__ v8f wmma4(v2f a, v2f b, v8f c) {
  // D = A(16x4) * B(4x16) + C   (fp32 WMMA, wave32)
  return __builtin_amdgcn_wmma_f32_16x16x4_f32(false, a, false, b, (short)0, c,
                                               false, false);
}

__device__ __forceinline__ float sigm(float x) { return 1.0f / (1.0f + __expf(-x)); }

__device__ __forceinline__ v8f splat8(float s) {
  v8f r = {s, s, s, s, s, s, s, s};
  return r;
}

// Workgroup = 2 waves, one 16-row batch tile.
//   wave 1 (producer): embedding gather + xw_t = x_t@k1 + b_in, one step AHEAD,
//                      double-buffered through LDS (its own SIMD/XDL pipe).
//   wave 0 (consumer): recurrent GEMMs (h@rk1), gates, full layer 2.
__global__ __launch_bounds__(64) void gru2_kernel(
    const int*   __restrict__ tokens,  // [B,S]
    const float* __restrict__ emb,     // [VOCAB,32]
    const float* __restrict__ k1,      // [32,96]
    const float* __restrict__ rk1,     // [32,96]
    const float* __restrict__ b1,      // [2,96]
    const float* __restrict__ k2,      // [32,48]
    const float* __restrict__ rk2,     // [16,48]
    const float* __restrict__ b2,      // [2,48]
    float*       __restrict__ out)     // [B,16]
{
  // xw handoff in C-fragment layout: same lane produces and consumes, flat copy.
  __shared__ float xw[2][6][32][8];  // [slot][ntile][lane][vgpr] = 12 KB
  __shared__ float h1buf[16][U1];    // layer-1 hidden, row-major (consumer only)
  __shared__ float h2buf[16][U2];    // layer-2 hidden, row-major (consumer only)

  const int tid  = threadIdx.x;
  const int wave = tid >> 5;
  const int lane = tid & 31;
  const int half = lane >> 4;
  const int n    = lane & 15;
  const int b0   = blockIdx.x * 16;

  if (wave == 1) {
    // ================= producer =================
    // B-operand fragments of k1: b[v] = k1[kt*4 + 2*half + v][j*16 + n]
    v2f k1f[8][6];
    #pragma unroll
    for (int k = 0; k < 8; ++k) {
      #pragma unroll
      for (int j = 0; j < 6; ++j) {
        const int r0 = k * 4 + half * 2;
        k1f[k][j].x = k1[(r0 + 0) * 96 + j * 16 + n];
        k1f[k][j].y = k1[(r0 + 1) * 96 + j * 16 + n];
      }
    }
    float bin1[6];
    #pragma unroll
    for (int j = 0; j < 6; ++j) bin1[j] = b1[j * 16 + n];

    // x A-fragments load directly from the embedding table:
    // lane needs row n, float pair at column k*4 + 2*half  (8-byte aligned).
    const int* tp = tokens + (long)(b0 + n) * SEQ;

    // software pipeline: xc = frags for t=p (ready), tokN = token for t=p+1.
    v2f xc[8];
    int tokN;
    {
      const int tok0 = tp[0];
      const float* e0 = emb + (long)tok0 * EMB;
      #pragma unroll
      for (int k = 0; k < 8; ++k) xc[k] = *(const v2f*)(e0 + k * 4 + half * 2);
      tokN = tp[1];
    }

    #pragma unroll 1
    for (int p = 0; p <= SEQ; ++p) {
      if (p < SEQ) {
        // issue prefetch first so it overlaps the WMMA chain
        v2f xn[8];
        {
          const float* en = emb + (long)tokN * EMB;
          #pragma unroll
          for (int k = 0; k < 8; ++k) xn[k] = *(const v2f*)(en + k * 4 + half * 2);
        }
        const int t2     = (p + 2 < SEQ) ? (p + 2) : (SEQ - 1);
        const int tokN2  = tp[t2];

        // xw(p) = x_p @ k1 + b_in   (48 WMMAs on this wave's XDL pipe)
        const int slot = p & 1;
        #pragma unroll
        for (int j = 0; j < 6; ++j) {
          v8f c = splat8(bin1[j]);
          #pragma unroll
          for (int k = 0; k < 8; ++k) c = wmma4(xc[k], k1f[k][j], c);
          float4* d = (float4*)&xw[slot][j][lane][0];
          float4 lo = {c[0], c[1], c[2], c[3]};
          float4 hi = {c[4], c[5], c[6], c[7]};
          d[0] = lo;
          d[1] = hi;
        }

        // rotate pipeline registers
        #pragma unroll
        for (int k = 0; k < 8; ++k) xc[k] = xn[k];
        tokN = tokN2;
      }
      __syncthreads();  // publishes xw(p); consumer finished slot (p-1)
    }
  } else {
    // ================= consumer =================
    v2f rk1f[8][6];
    #pragma unroll
    for (int k = 0; k < 8; ++k) {
      #pragma unroll
      for (int j = 0; j < 6; ++j) {
        const int r0 = k * 4 + half * 2;
        rk1f[k][j].x = rk1[(r0 + 0) * 96 + j * 16 + n];
        rk1f[k][j].y = rk1[(r0 + 1) * 96 + j * 16 + n];
      }
    }
    v2f k2f[8][3];
    #pragma unroll
    for (int k = 0; k < 8; ++k) {
      #pragma unroll
      for (int j = 0; j < 3; ++j) {
        const int r0 = k * 4 + half * 2;
        k2f[k][j].x = k2[(r0 + 0) * 48 + j * 16 + n];
        k2f[k][j].y = k2[(r0 + 1) * 48 + j * 16 + n];
      }
    }
    v2f rk2f[4][3];
    #pragma unroll
    for (int k = 0; k < 4; ++k) {
      #pragma unroll
      for (int j = 0; j < 3; ++j) {
        const int r0 = k * 4 + half * 2;
        rk2f[k][j].x = rk2[(r0 + 0) * 48 + j * 16 + n];
        rk2f[k][j].y = rk2[(r0 + 1) * 48 + j * 16 + n];
      }
    }
    float brec1[6];
    #pragma unroll
    for (int j = 0; j < 6; ++j) brec1[j] = b1[96 + j * 16 + n];
    float bin2[3], brec2[3];
    #pragma unroll
    for (int j = 0; j < 3; ++j) {
      bin2[j]  = b2[j * 16 + n];
      brec2[j] = b2[48 + j * 16 + n];
    }

    v8f h1c[2];
    h1c[0] = splat8(0.0f);
    h1c[1] = splat8(0.0f);
    v8f h2c = splat8(0.0f);
    for (int i = lane; i < 16 * U1; i += 32) (&h1buf[0][0])[i] = 0.0f;
    for (int i = lane; i < 16 * U2; i += 32) (&h2buf[0][0])[i] = 0.0f;

    #pragma unroll 1
    for (int p = 0; p <= SEQ; ++p) {
      if (p > 0) {
        const int slot = (p - 1) & 1;

        // xacc handed over from producer (lane-identical C fragments)
        v8f xacc[6];
        #pragma unroll
        for (int j = 0; j < 6; ++j) {
          const float4* s = (const float4*)&xw[slot][j][lane][0];
          float4 lo = s[0], hi = s[1];
          v8f c = {lo.x, lo.y, lo.z, lo.w, hi.x, hi.y, hi.z, hi.w};
          xacc[j] = c;
        }

        // h1 A-fragments from LDS: a[v] = h1[row=n][kt*4 + 2*half + v]
        v2f ha[8];
        #pragma unroll
        for (int k = 0; k < 8; ++k) {
          const float* p1 = &h1buf[n][k * 4 + half * 2];
          ha[k].x = p1[0];
          ha[k].y = p1[1];
        }

        // rec = h @ rk1 + b_rec  (48 WMMAs — the recurrent critical path)
        v8f racc[6];
        #pragma unroll
        for (int j = 0; j < 6; ++j) {
          v8f rc = splat8(brec1[j]);
          #pragma unroll
          for (int k = 0; k < 8; ++k) rc = wmma4(ha[k], rk1f[k][j], rc);
          racc[j] = rc;
        }

        // gates (z | r | h thirds of the 96-wide projection)
        #pragma unroll
        for (int j = 0; j < 2; ++j) {
          v8f hn;
          #pragma unroll
          for (int v = 0; v < 8; ++v) {
            const float z  = sigm(xacc[j][v]     + racc[j][v]);
            const float r  = sigm(xacc[2 + j][v] + racc[2 + j][v]);
            const float hh = sigm(xacc[4 + j][v] + r * racc[4 + j][v]);
            hn[v] = z * h1c[j][v] + (1.0f - z) * hh;
          }
          h1c[j] = hn;
          #pragma unroll
          for (int v = 0; v < 8; ++v)
            h1buf[v + 8 * half][j * 16 + n] = hn[v];  // C-layout scatter
        }

        // layer 2 A-fragments (DS in-order within wave: sees fresh h1buf)
        v2f s1a[8], h2a[4];
        #pragma unroll
        for (int k = 0; k < 8; ++k) {
          const float* p1 = &h1buf[n][k * 4 + half * 2];
          s1a[k].x = p1[0];
          s1a[k].y = p1[1];
        }
        #pragma unroll
        for (int k = 0; k < 4; ++k) {
          const float* p2 = &h2buf[n][k * 4 + half * 2];
          h2a[k].x = p2[0];
          h2a[k].y = p2[1];
        }

        // layer 2: xw2 = s1@k2 + b ; rec2 = h2@rk2 + b   (36 WMMAs)
        v8f x2[3], r2[3];
        #pragma unroll
        for (int j = 0; j < 3; ++j) {
          v8f c1 = splat8(bin2[j]);
          #pragma unroll
          for (int k = 0; k < 8; ++k) c1 = wmma4(s1a[k], k2f[k][j], c1);
          x2[j] = c1;
          v8f c2 = splat8(brec2[j]);
          #pragma unroll
          for (int k = 0; k < 4; ++k) c2 = wmma4(h2a[k], rk2f[k][j], c2);
          r2[j] = c2;
        }

        v8f h2n;
        #pragma unroll
        for (int v = 0; v < 8; ++v) {
          const float z  = sigm(x2[0][v] + r2[0][v]);
          const float r  = sigm(x2[1][v] + r2[1][v]);
          const float hh = sigm(x2[2][v] + r * r2[2][v]);
          h2n[v] = z * h2c[v] + (1.0f - z) * hh;
        }
        h2c = h2n;
        #pragma unroll
        for (int v = 0; v < 8; ++v)
          h2buf[v + 8 * half][n] = h2n[v];
      }
      __syncthreads();
    }

    // final layer-2 hidden state -> out [B,16]
    #pragma unroll
    for (int v = 0; v < 8; ++v)
      out[(long)(b0 + v + 8 * half) * U2 + n] = h2c[v];
  }
}

extern "C" void kernel_launch(void* const* d_in, const int* in_sizes, int n_in,
                              void* d_out, int out_size, void* d_ws, size_t ws_size,
                              hipStream_t stream) {
  const int*   tokens = (const int*)  d_in[0];
  const float* emb    = (const float*)d_in[1];
  const float* k1     = (const float*)d_in[2];
  const float* rk1    = (const float*)d_in[3];
  const float* b1     = (const float*)d_in[4];
  const float* k2     = (const float*)d_in[5];
  const float* rk2    = (const float*)d_in[6];
  const float* b2     = (const float*)d_in[7];
  float* out = (float*)d_out;

  dim3 grid(BATCH / 16);
  dim3 block(64);  // 2 waves: producer + consumer
  gru2_kernel<<<grid, block, 0, stream>>>(tokens, emb, k1, rk1, b1, k2, rk2, b2, out);
}